// DeformableTransformerDecoderLayer_Det_54683523612975
// MI455X (gfx1250) — compile-verified
//
#include <hip/hip_runtime.h>
#include <math.h>
#include <stdint.h>

// ---- problem constants (match reference) ----
#define B_    4
#define Q_    300
#define P_    16
#define C_    256
#define NH_   8
#define DH_   32
#define NL_   4
#define NPT_  4
#define LIN_  21760
#define DFF_  1024
#define M1_   (B_*Q_*P_)    // 19200 query rows
#define MV_   (B_*LIN_)     // 87040 value rows

typedef __attribute__((ext_vector_type(2))) float v2f;
typedef __attribute__((ext_vector_type(8))) float v8f;

// ---------------------------------------------------------------------------
// CDNA5 async memory->LDS DMA (ASYNCcnt tracked), 16 bytes per lane.
// LDS destination = low 32 bits of the generic shared-memory pointer
// (LDS aperture places the byte offset in addr[31:0]).
// ---------------------------------------------------------------------------
__device__ __forceinline__ void async_ld_b128(void* lds_dst, const float* gaddr) {
  const unsigned lds_addr = (unsigned)(uintptr_t)lds_dst;
  asm volatile("global_load_async_to_lds_b128 %0, %1, off"
               :: "v"(lds_addr), "v"(gaddr)
               : "memory");
}
__device__ __forceinline__ void wait_async_zero() {
  asm volatile("s_wait_asynccnt 0x0" ::: "memory");
}

// ---------------------------------------------------------------------------
// Generic fp32 WMMA GEMM:  D[M,N] = A[M,K] @ W[N,K]^T + bias[N]   (opt ReLU)
// One wave (32 threads) computes a 16x64 output tile using
// v_wmma_f32_16x16x4_f32, K stepped by 4, software-pipelined depth 2.
// The k-loop is kept rolled so the shift-register pipeline survives regalloc:
// WMMAs of step k wait only on loads issued two iterations earlier
// (s_wait_loadcnt <= 10), hiding ~2 iterations of VMEM latency per wave,
// with single-wave blocks providing additional wave-level overlap.
// A-operand per lane: A[m0 + lane%16][k + 2*(lane>>4) + {0,1}]
// B-operand per lane: W[n0 + lane%16][k + 2*(lane>>4) + {0,1}]  (B = W^T)
// C/D layout: vgpr rr, lane L -> row m0 + rr + 8*(L>>4), col n0 + (L&15)
// Requires M%16==0, N%64==0, K%4==0, K>=8.
// ---------------------------------------------------------------------------
template<int ACT>
__global__ __launch_bounds__(32)
void k_gemm(const float* __restrict__ A, const float* __restrict__ W,
            const float* __restrict__ bias, float* __restrict__ D,
            int M, int N, int K) {
  const int m0   = blockIdx.x * 16;
  const int n0   = blockIdx.y * 64;
  const int lane = threadIdx.x;
  const int half = lane >> 4;
  const int r    = lane & 15;

  v8f acc0 = {}, acc1 = {}, acc2 = {}, acc3 = {};

  const float* Ap = A + (size_t)(m0 + r) * K + 2 * half;
  const float* W0 = W + (size_t)(n0 + r) * K + 2 * half;
  const float* W1 = W0 + (size_t)16 * K;
  const float* W2 = W0 + (size_t)32 * K;
  const float* W3 = W0 + (size_t)48 * K;

  // bias loads issued early; consumed only after the k-loop
  const float bb0 = bias[n0 + r];
  const float bb1 = bias[n0 + 16 + r];
  const float bb2 = bias[n0 + 32 + r];
  const float bb3 = bias[n0 + 48 + r];

  // stage 0 fragments (k = 0)
  v2f a_p0  = *(const v2f*)(Ap);
  v2f b0_p0 = *(const v2f*)(W0);
  v2f b1_p0 = *(const v2f*)(W1);
  v2f b2_p0 = *(const v2f*)(W2);
  v2f b3_p0 = *(const v2f*)(W3);
  // stage 1 fragments (k = 4)
  v2f a_p1  = *(const v2f*)(Ap + 4);
  v2f b0_p1 = *(const v2f*)(W0 + 4);
  v2f b1_p1 = *(const v2f*)(W1 + 4);
  v2f b2_p1 = *(const v2f*)(W2 + 4);
  v2f b3_p1 = *(const v2f*)(W3 + 4);

#pragma clang loop unroll(disable)
  for (int k = 8; k < K; k += 4) {
    // issue loads for step k (stay in flight across the next two WMMA groups)
    v2f a_n  = *(const v2f*)(Ap + k);
    v2f b0_n = *(const v2f*)(W0 + k);
    v2f b1_n = *(const v2f*)(W1 + k);
    v2f b2_n = *(const v2f*)(W2 + k);
    v2f b3_n = *(const v2f*)(W3 + k);
    // consume step k-8 fragments (loaded two iterations ago)
    acc0 = __builtin_amdgcn_wmma_f32_16x16x4_f32(false, a_p0, false, b0_p0, (short)0, acc0, false, false);
    acc1 = __builtin_amdgcn_wmma_f32_16x16x4_f32(false, a_p0, false, b1_p0, (short)0, acc1, false, false);
    acc2 = __builtin_amdgcn_wmma_f32_16x16x4_f32(false, a_p0, false, b2_p0, (short)0, acc2, false, false);
    acc3 = __builtin_amdgcn_wmma_f32_16x16x4_f32(false, a_p0, false, b3_p0, (short)0, acc3, false, false);
    // shift pipeline
    a_p0 = a_p1;  b0_p0 = b0_p1; b1_p0 = b1_p1; b2_p0 = b2_p1; b3_p0 = b3_p1;
    a_p1 = a_n;   b0_p1 = b0_n;  b1_p1 = b1_n;  b2_p1 = b2_n;  b3_p1 = b3_n;
  }
  // drain the two in-flight stages
  acc0 = __builtin_amdgcn_wmma_f32_16x16x4_f32(false, a_p0, false, b0_p0, (short)0, acc0, false, false);
  acc1 = __builtin_amdgcn_wmma_f32_16x16x4_f32(false, a_p0, false, b1_p0, (short)0, acc1, false, false);
  acc2 = __builtin_amdgcn_wmma_f32_16x16x4_f32(false, a_p0, false, b2_p0, (short)0, acc2, false, false);
  acc3 = __builtin_amdgcn_wmma_f32_16x16x4_f32(false, a_p0, false, b3_p0, (short)0, acc3, false, false);
  acc0 = __builtin_amdgcn_wmma_f32_16x16x4_f32(false, a_p1, false, b0_p1, (short)0, acc0, false, false);
  acc1 = __builtin_amdgcn_wmma_f32_16x16x4_f32(false, a_p1, false, b1_p1, (short)0, acc1, false, false);
  acc2 = __builtin_amdgcn_wmma_f32_16x16x4_f32(false, a_p1, false, b2_p1, (short)0, acc2, false, false);
  acc3 = __builtin_amdgcn_wmma_f32_16x16x4_f32(false, a_p1, false, b3_p1, (short)0, acc3, false, false);

#pragma unroll
  for (int rr = 0; rr < 8; ++rr) {
    const int m = m0 + rr + 8 * half;
    float* drow = D + (size_t)m * N + n0 + r;
    float o0 = acc0[rr] + bb0;
    float o1 = acc1[rr] + bb1;
    float o2 = acc2[rr] + bb2;
    float o3 = acc3[rr] + bb3;
    if (ACT) { o0 = fmaxf(o0, 0.f); o1 = fmaxf(o1, 0.f); o2 = fmaxf(o2, 0.f); o3 = fmaxf(o3, 0.f); }
    drow[0]  = o0;
    drow[16] = o1;
    drow[32] = o2;
    drow[48] = o3;
  }
}

// ---------------------------------------------------------------------------
// tgt (B,Q,P,C) -> x (B,P,Q,C)
// ---------------------------------------------------------------------------
__global__ __launch_bounds__(256)
void k_transpose_tgt(const float* __restrict__ tgt, float* __restrict__ x) {
  const int row = blockIdx.x;               // (b,p,q) flat
  const int b = row / (P_ * Q_);
  const int rem = row % (P_ * Q_);
  const int p = rem / Q_;
  const int q = rem % Q_;
  const int srow = (b * Q_ + q) * P_ + p;
  x[(size_t)row * C_ + threadIdx.x] = tgt[(size_t)srow * C_ + threadIdx.x];
}

// ---------------------------------------------------------------------------
// block-wide sum over 256 threads
// ---------------------------------------------------------------------------
__device__ __forceinline__ float blk_sum(float v, float* red) {
  const int t = threadIdx.x;
  red[t] = v; __syncthreads();
#pragma unroll
  for (int s = 128; s > 0; s >>= 1) {
    if (t < s) red[t] += red[t + s];
    __syncthreads();
  }
  float r = red[0];
  __syncthreads();
  return r;
}

// ---------------------------------------------------------------------------
// Self-attention: one block per (n, h); K/V tiles DMAed into LDS with
// global_load_async_to_lds_b128 (ASYNCcnt), no VGPR round-trip.
// qkv layout: [(n*Q + q)*768 + {0|256|512} + h*32 + d].  dst layout (B,P,Q,C).
// ---------------------------------------------------------------------------
__global__ __launch_bounds__(256)
void k_attn(const float* __restrict__ qkv, float* __restrict__ dst) {
  const int n = blockIdx.x >> 3;   // / NH_
  const int h = blockIdx.x & 7;
  extern __shared__ float sm[];
  float* sK  = sm;                  // 300*32
  float* sV  = sK + 300 * 32;       // 300*32
  float* sQv = sV + 300 * 32;       // 32
  float* sS  = sQv + 32;            // 304
  float* red = sS + 304;            // 256
  const int tid = threadIdx.x;
  const size_t baseN = (size_t)n * Q_ * 768;

  // Stage K and V (300x32 each) via async DMA: 16B chunks, fully aligned.
  for (int i = tid; i < (300 * 32) / 4; i += 256) {      // 2400 float4 chunks
    const int kq = i >> 3, seg = i & 7;                  // 8 chunks per row
    const float* gk = qkv + baseN + (size_t)kq * 768 + 256 + h * 32 + seg * 4;
    async_ld_b128(sK + i * 4, gk);
    async_ld_b128(sV + i * 4, gk + 256);
  }
  wait_async_zero();
  __syncthreads();

  const float scale = 0.17677669529663687f;   // 1/sqrt(32)
  for (int q = 0; q < Q_; ++q) {
    if (tid < 32) sQv[tid] = qkv[baseN + (size_t)q * 768 + h * 32 + tid];
    __syncthreads();

    float pm = -3.4e38f;
    for (int k = tid; k < Q_; k += 256) {
      float s = 0.f;
#pragma unroll 8
      for (int d = 0; d < 32; ++d) s += sQv[d] * sK[k * 32 + d];
      s *= scale;
      sS[k] = s;
      pm = fmaxf(pm, s);
    }
    red[tid] = pm; __syncthreads();
#pragma unroll
    for (int s2 = 128; s2 > 0; s2 >>= 1) {
      if (tid < s2) red[tid] = fmaxf(red[tid], red[tid + s2]);
      __syncthreads();
    }
    const float mx = red[0]; __syncthreads();

    float ps = 0.f;
    for (int k = tid; k < Q_; k += 256) {
      const float e = __expf(sS[k] - mx);
      sS[k] = e;
      ps += e;
    }
    const float ssum = blk_sum(ps, red);
    const float inv = 1.f / ssum;

    const int d = tid & 31, g = tid >> 5;
    float acc = 0.f;
    for (int k = g; k < Q_; k += 8) acc += sS[k] * sV[k * 32 + d];
    red[tid] = acc; __syncthreads();
    if (tid < 32) {
      const float s3 = red[tid] + red[tid + 32] + red[tid + 64] + red[tid + 96] +
                       red[tid + 128] + red[tid + 160] + red[tid + 192] + red[tid + 224];
      dst[((size_t)n * Q_ + q) * C_ + h * 32 + tid] = s3 * inv;
    }
    __syncthreads();
  }
}

// ---------------------------------------------------------------------------
// LN( x + o ) over C  in (B,P,Q,C) layout -> x1; fused query = ln + query_pos
// written in (B,Q,P,C) layout.
// ---------------------------------------------------------------------------
__global__ __launch_bounds__(256)
void k_ln_inter(const float* __restrict__ x, const float* __restrict__ o,
                const float* __restrict__ g, const float* __restrict__ be,
                const float* __restrict__ qpos,
                float* __restrict__ x1, float* __restrict__ query) {
  __shared__ float red[256];
  const int row = blockIdx.x;       // (b,p,q)
  const int c = threadIdx.x;
  const float v = x[(size_t)row * C_ + c] + o[(size_t)row * C_ + c];
  const float m = blk_sum(v, red) * (1.f / C_);
  const float dv = v - m;
  const float var = blk_sum(dv * dv, red) * (1.f / C_);
  const float ln = dv * rsqrtf(var + 1e-5f) * g[c] + be[c];
  x1[(size_t)row * C_ + c] = ln;
  const int b = row / (P_ * Q_);
  const int rem = row % (P_ * Q_);
  const int p = rem / Q_;
  const int q = rem % Q_;
  const size_t row2 = (size_t)(b * Q_ + q) * P_ + p;   // (b,q,p)
  query[row2 * C_ + c] = ln + qpos[row2 * C_ + c];
}

// ---------------------------------------------------------------------------
// x2 = LN( x1[(b,p,q)] + t2[(b,q,p)] )  written in (B,Q,P,C) layout.
// ---------------------------------------------------------------------------
__global__ __launch_bounds__(256)
void k_ln_cross(const float* __restrict__ x1, const float* __restrict__ t2,
                const float* __restrict__ g, const float* __restrict__ be,
                float* __restrict__ x2) {
  __shared__ float red[256];
  const int row2 = blockIdx.x;      // (b,q,p)
  const int c = threadIdx.x;
  const int b = row2 / (Q_ * P_);
  const int rem = row2 % (Q_ * P_);
  const int q = rem / P_;
  const int p = rem % P_;
  const size_t rowx = (size_t)(b * P_ + p) * Q_ + q;
  const float v = x1[rowx * C_ + c] + t2[(size_t)row2 * C_ + c];
  const float m = blk_sum(v, red) * (1.f / C_);
  const float dv = v - m;
  const float var = blk_sum(dv * dv, red) * (1.f / C_);
  x2[(size_t)row2 * C_ + c] = dv * rsqrtf(var + 1e-5f) * g[c] + be[c];
}

// ---------------------------------------------------------------------------
// out = LN( a + bres ), same-layout rows.
// ---------------------------------------------------------------------------
__global__ __launch_bounds__(256)
void k_ln3(const float* __restrict__ a, const float* __restrict__ bres,
           const float* __restrict__ g, const float* __restrict__ be,
           float* __restrict__ out) {
  __shared__ float red[256];
  const int row = blockIdx.x;
  const int c = threadIdx.x;
  const float v = a[(size_t)row * C_ + c] + bres[(size_t)row * C_ + c];
  const float m = blk_sum(v, red) * (1.f / C_);
  const float dv = v - m;
  const float var = blk_sum(dv * dv, red) * (1.f / C_);
  out[(size_t)row * C_ + c] = dv * rsqrtf(var + 1e-5f) * g[c] + be[c];
}

// ---------------------------------------------------------------------------
// in-place softmax over groups of 16 (attention weights per (row, head))
// ---------------------------------------------------------------------------
__global__ __launch_bounds__(256)
void k_softmax16(float* __restrict__ aw, int ngroups) {
  const int gi = blockIdx.x * blockDim.x + threadIdx.x;
  if (gi >= ngroups) return;
  float* p = aw + (size_t)gi * 16;
  float mx = p[0];
#pragma unroll
  for (int i = 1; i < 16; ++i) mx = fmaxf(mx, p[i]);
  float e[16]; float s = 0.f;
#pragma unroll
  for (int i = 0; i < 16; ++i) { e[i] = __expf(p[i] - mx); s += e[i]; }
  const float inv = 1.f / s;
#pragma unroll
  for (int i = 0; i < 16; ++i) p[i] = e[i] * inv;
}

// ---------------------------------------------------------------------------
// Multi-scale deformable sampling.  block = query row (b,q,p); thread = channel
// c = h*32+d.  All 32 lanes of a head share the 4 bilinear tap addresses.
// ---------------------------------------------------------------------------
__global__ __launch_bounds__(256)
void k_msda(const float* __restrict__ val, const float* __restrict__ ref,
            const float* __restrict__ off, const float* __restrict__ aw,
            float* __restrict__ out) {
  const int row2 = blockIdx.x;              // b*Q*P + q*P + p  (== b*Lq + lq)
  const int b = row2 / (Q_ * P_);
  const int c = threadIdx.x;
  const int h = c >> 5, d = c & 31;
  const int Hs[4] = {128, 64, 32, 16};
  const int Ws[4] = {128, 64, 32, 16};
  const int St[4] = {0, 16384, 20480, 21504};

  const float* refp = ref + (size_t)row2 * (NL_ * 2);
  const float* offp = off + (size_t)row2 * C_;
  const float* awp  = aw  + (size_t)row2 * (NH_ * NL_ * NPT_);

  float acc = 0.f;
#pragma unroll
  for (int l = 0; l < NL_; ++l) {
    const int H = Hs[l], W = Ws[l];
    const float* vb = val + ((size_t)b * LIN_ + St[l]) * C_ + h * DH_ + d;
    const float rx = refp[l * 2 + 0];
    const float ry = refp[l * 2 + 1];
#pragma unroll
    for (int pt = 0; pt < NPT_; ++pt) {
      const int oi = (h * NL_ + l) * NPT_ + pt;   // == h*16 + l*4 + pt
      const float ox = offp[oi * 2 + 0];
      const float oy = offp[oi * 2 + 1];
      const float a  = awp[oi];
      const float lx = rx + ox / (float)W;
      const float ly = ry + oy / (float)H;
      const float fx = lx * (float)W - 0.5f;
      const float fy = ly * (float)H - 0.5f;
      const float x0f = floorf(fx), y0f = floorf(fy);
      const float wx = fx - x0f, wy = fy - y0f;
      const int x0 = (int)x0f, y0 = (int)y0f;
      const int x1 = x0 + 1,  y1 = y0 + 1;
      float s = 0.f;
      if (x0 >= 0 && x0 < W && y0 >= 0 && y0 < H) { const int idx = y0 * W + x0; s += vb[(size_t)idx * C_] * (1.f - wx) * (1.f - wy); }
      if (x1 >= 0 && x1 < W && y0 >= 0 && y0 < H) { const int idx = y0 * W + x1; s += vb[(size_t)idx * C_] * wx * (1.f - wy); }
      if (x0 >= 0 && x0 < W && y1 >= 0 && y1 < H) { const int idx = y1 * W + x0; s += vb[(size_t)idx * C_] * (1.f - wx) * wy; }
      if (x1 >= 0 && x1 < W && y1 >= 0 && y1 < H) { const int idx = y1 * W + x1; s += vb[(size_t)idx * C_] * wx * wy; }
      acc += a * s;
    }
  }
  out[(size_t)row2 * C_ + c] = acc;
}

// ---------------------------------------------------------------------------
extern "C" void kernel_launch(void* const* d_in, const int* in_sizes, int n_in,
                              void* d_out, int out_size, void* d_ws, size_t ws_size,
                              hipStream_t stream) {
  const float* tgt        = (const float*)d_in[0];
  const float* qpos       = (const float*)d_in[1];
  const float* refpts     = (const float*)d_in[2];
  const float* src        = (const float*)d_in[3];
  // d_in[4] spatial shapes, d_in[5] level starts: compile-time constants here
  const float* inproj_w   = (const float*)d_in[6];
  const float* inproj_b   = (const float*)d_in[7];
  const float* outproj_w  = (const float*)d_in[8];
  const float* outproj_b  = (const float*)d_in[9];
  const float* ln_inter_g = (const float*)d_in[10];
  const float* ln_inter_b = (const float*)d_in[11];
  const float* ln_cross_g = (const float*)d_in[12];
  const float* ln_cross_b = (const float*)d_in[13];
  const float* ln3_g      = (const float*)d_in[14];
  const float* ln3_b      = (const float*)d_in[15];
  const float* off_w      = (const float*)d_in[16];
  const float* off_b      = (const float*)d_in[17];
  const float* attw_w     = (const float*)d_in[18];
  const float* attw_b     = (const float*)d_in[19];
  const float* val_w      = (const float*)d_in[20];
  const float* val_b      = (const float*)d_in[21];
  const float* mout_w     = (const float*)d_in[22];
  const float* mout_b     = (const float*)d_in[23];
  const float* lin1_w     = (const float*)d_in[24];
  const float* lin1_b     = (const float*)d_in[25];
  const float* lin2_w     = (const float*)d_in[26];
  const float* lin2_b     = (const float*)d_in[27];
  float* out = (float*)d_out;

  // ---- workspace layout (floats), with deterministic region reuse ----
  float* ws = (float*)d_ws;
  const size_t SZ_X   = (size_t)M1_ * C_;       //  4,915,200
  const size_t SZ_QKV = (size_t)M1_ * 3 * C_;   // 14,745,600
  const size_t SZ_VAL = (size_t)MV_ * C_;       // 22,282,240
  float* X   = ws;                 // x (B,P,Q,C); later: query / msda_raw / ffn2
  float* QKV = X + SZ_X;           // qkv;        later: off / aw / t2
  float* O   = QKV + SZ_QKV;       // attn outproj; later: x2
  float* X1  = O + SZ_X;           // attn head out, then ln_inter out
  float* VAL = X1 + SZ_X;          // value proj;  later: ffn hidden
  (void)SZ_VAL; (void)ws_size; (void)in_sizes; (void)n_in; (void)out_size;

  float* QUERY = X;
  float* OFF   = QKV;
  float* AW    = QKV + SZ_X;
  float* T2    = AW + (size_t)M1_ * NH_ * NL_ * NPT_;
  float* X2    = O;
  float* HB    = VAL;
  float* MS    = X;
  float* F2    = X;

  // 1) tgt -> x (B,P,Q,C)
  k_transpose_tgt<<<M1_, 256, 0, stream>>>(tgt, X);
  // 2) qkv = x @ inproj^T + b        (19200 x 768, K=256)
  k_gemm<0><<<dim3(M1_ / 16, (3 * C_) / 64), 32, 0, stream>>>(X, inproj_w, inproj_b, QKV, M1_, 3 * C_, C_);
  // 3) self-attention -> X1 (head-concat layout (B,P,Q,C))
  const size_t attsm = (size_t)(300 * 32 * 2 + 32 + 304 + 256) * sizeof(float);
  k_attn<<<B_ * P_ * NH_, 256, attsm, stream>>>(QKV, X1);
  // 4) out-projection -> O
  k_gemm<0><<<dim3(M1_ / 16, C_ / 64), 32, 0, stream>>>(X1, outproj_w, outproj_b, O, M1_, C_, C_);
  // 5) value projection of src -> VAL (87040 x 256)
  k_gemm<0><<<dim3(MV_ / 16, C_ / 64), 32, 0, stream>>>(src, val_w, val_b, VAL, MV_, C_, C_);
  // 6) x1 = LN(x + o); query = x1 + query_pos (transposed write)
  k_ln_inter<<<M1_, 256, 0, stream>>>(X, O, ln_inter_g, ln_inter_b, qpos, X1, QUERY);
  // 7) sampling offsets  (19200 x 256)
  k_gemm<0><<<dim3(M1_ / 16, C_ / 64), 32, 0, stream>>>(QUERY, off_w, off_b, OFF, M1_, C_, C_);
  // 8) attention weights (19200 x 128)
  k_gemm<0><<<dim3(M1_ / 16, 128 / 64), 32, 0, stream>>>(QUERY, attw_w, attw_b, AW, M1_, 128, C_);
  // 9) softmax over NL*NP=16 per (row, head)
  k_softmax16<<<(M1_ * NH_ + 255) / 256, 256, 0, stream>>>(AW, M1_ * NH_);
  // 10) deformable bilinear sampling -> MS (overwrites QUERY region; query dead)
  k_msda<<<M1_, 256, 0, stream>>>(VAL, refpts, OFF, AW, MS);
  // 11) msda output projection -> T2
  k_gemm<0><<<dim3(M1_ / 16, C_ / 64), 32, 0, stream>>>(MS, mout_w, mout_b, T2, M1_, C_, C_);
  // 12) x2 = LN(x1 + t2)  in (B,Q,P,C)
  k_ln_cross<<<M1_, 256, 0, stream>>>(X1, T2, ln_cross_g, ln_cross_b, X2);
  // 13) ffn hidden = relu(x2 @ lin1^T + b1)  (19200 x 1024)
  k_gemm<1><<<dim3(M1_ / 16, DFF_ / 64), 32, 0, stream>>>(X2, lin1_w, lin1_b, HB, M1_, DFF_, C_);
  // 14) ffn out = h @ lin2^T + b2  (19200 x 256, K=1024)
  k_gemm<0><<<dim3(M1_ / 16, C_ / 64), 32, 0, stream>>>(HB, lin2_w, lin2_b, F2, M1_, C_, DFF_);
  // 15) out = LN(x2 + ffn_out)
  k_ln3<<<M1_, 256, 0, stream>>>(X2, F2, ln3_g, ln3_b, out);
}